// BatchHardTripletSelector_53944789237989
// MI455X (gfx1250) — compile-verified
//
#include <hip/hip_runtime.h>
#include <hip/hip_bf16.h>

typedef float v2f __attribute__((ext_vector_type(2)));
typedef float v8f __attribute__((ext_vector_type(8)));

#define NN 8192
#define DD 256
#define ROWS_PER_BLOCK 128
#define COL_SPLITS 4
#define COLS_PER_SPLIT (NN / COL_SPLITS)        /* 2048 */
#define TILES_PER_SPLIT (COLS_PER_SPLIT / 16)   /* 128  */
#define LDSP (DD + 4)                           /* 260 floats: 16B-aligned rows, bank stride 4 */

#if defined(__has_builtin)
#if __has_builtin(__builtin_amdgcn_global_load_async_to_lds_b128) && \
    __has_builtin(__builtin_amdgcn_s_wait_asynccnt)
#define HAS_ASYNC_LDS 1
#endif
#endif
#ifndef HAS_ASYNC_LDS
#define HAS_ASYNC_LDS 0
#endif

#if HAS_ASYNC_LDS
// Builtin signature (from hipcc diagnostic): 4 args, ptr-to-int-vector(4).
typedef int v4i __attribute__((vector_size(16)));
typedef __attribute__((address_space(1))) v4i g_v4i;   // global
typedef __attribute__((address_space(3))) v4i l_v4i;   // LDS

// Memory -> LDS, 16 bytes, no VGPR round trip; tracked by ASYNCcnt.
// Address-space handling via integer casts: generic->AS1 keeps the 64-bit
// address; generic->AS3 takes addr[31:0], which is the ISA's flat->LDS map.
__device__ __forceinline__ void async_cp16(const float* gsrc, float* ldst) {
  __builtin_amdgcn_global_load_async_to_lds_b128(
      (g_v4i*)((uintptr_t)gsrc),
      (l_v4i*)(unsigned)((uintptr_t)ldst),
      0, 0);
}
#endif

// ---------------------------------------------------------------------------
// Kernel 1: sq[j] = ||embeds[j]||^2   (one wave32 per row)
// ---------------------------------------------------------------------------
__global__ __launch_bounds__(256) void tk_sqnorm(const float* __restrict__ e,
                                                 float* __restrict__ sq) {
  const int wave = threadIdx.x >> 5, lane = threadIdx.x & 31;
  const int row = blockIdx.x * 8 + wave;
  const float* p = e + row * DD + lane;
  float s = 0.f;
#pragma unroll
  for (int q = 0; q < DD / 32; ++q) { float v = p[q * 32]; s += v * v; }
#pragma unroll
  for (int m = 16; m >= 1; m >>= 1) s += __shfl_xor(s, m, 32);
  if (lane == 0) sq[row] = s;
}

// ---------------------------------------------------------------------------
// Kernel 2: fused fp32 WMMA GEMM + batch-hard argmax/argmin epilogue.
// grid = (N/128, 4 column splits), block = 256 (8 waves; wave w owns 16 rows).
// B tiles double-buffered: next tile's async global->LDS transfer overlaps
// the current tile's 64 WMMAs; one barrier per tile.
// ---------------------------------------------------------------------------
__global__ __launch_bounds__(256) void tk_hardmine(
    const float* __restrict__ e, const int* __restrict__ labels,
    const float* __restrict__ sq,
    float* __restrict__ posv, int* __restrict__ posi,
    float* __restrict__ negv, int* __restrict__ negi) {
  extern __shared__ float lds[];
  float* As = lds;                               // [128][LDSP]
  float* Bs = lds + ROWS_PER_BLOCK * LDSP;       // 2 x [16][LDSP]

  const int t = threadIdx.x;
  const int wave = t >> 5, lane = t & 31;
  const int l16 = lane & 15, lhalf = lane >> 4;
  const int rowBase = blockIdx.x * ROWS_PER_BLOCK;
  const int split = blockIdx.y;
  const int colBase0 = split * COLS_PER_SPLIT;
  const float INF = __builtin_inff();

  // Per-thread share of a 16x256 tile: 4 float4s at (r, c4).
  const int sr0 = (t + 0 * 256) >> 6, sc0 = (t + 0 * 256) & 63;
  const int sr1 = (t + 1 * 256) >> 6, sc1 = (t + 1 * 256) & 63;
  const int sr2 = (t + 2 * 256) >> 6, sc2 = (t + 2 * 256) & 63;
  const int sr3 = (t + 3 * 256) >> 6, sc3 = (t + 3 * 256) & 63;

  // ---- Stage the 128-row A block into LDS once ----
#if HAS_ASYNC_LDS
#pragma unroll
  for (int it = 0; it < (ROWS_PER_BLOCK * (DD / 4)) / 256; ++it) {
    const int idx = t + it * 256;
    const int r = idx >> 6, c4 = idx & 63;
    async_cp16(e + (size_t)(rowBase + r) * DD + (c4 << 2), As + r * LDSP + (c4 << 2));
  }
#else
#pragma unroll
  for (int it = 0; it < (ROWS_PER_BLOCK * (DD / 4)) / 256; it += 4) {
    float4 a0, a1, a2, a3;
    {
      const int i0 = t + (it + 0) * 256, i1 = t + (it + 1) * 256;
      const int i2 = t + (it + 2) * 256, i3 = t + (it + 3) * 256;
      const float4* g = (const float4*)(e + (size_t)rowBase * DD);
      a0 = g[(i0 >> 6) * (DD / 4) + (i0 & 63)];
      a1 = g[(i1 >> 6) * (DD / 4) + (i1 & 63)];
      a2 = g[(i2 >> 6) * (DD / 4) + (i2 & 63)];
      a3 = g[(i3 >> 6) * (DD / 4) + (i3 & 63)];
      *(float4*)(As + (i0 >> 6) * LDSP + ((i0 & 63) << 2)) = a0;
      *(float4*)(As + (i1 >> 6) * LDSP + ((i1 & 63) << 2)) = a1;
      *(float4*)(As + (i2 >> 6) * LDSP + ((i2 & 63) << 2)) = a2;
      *(float4*)(As + (i3 >> 6) * LDSP + ((i3 & 63) << 2)) = a3;
    }
  }
#endif

  // ---- Stage B tile 0 into buffer 0 ----
  {
    const float* g0 = e + (size_t)colBase0 * DD;
#if HAS_ASYNC_LDS
    async_cp16(g0 + sr0 * DD + (sc0 << 2), Bs + sr0 * LDSP + (sc0 << 2));
    async_cp16(g0 + sr1 * DD + (sc1 << 2), Bs + sr1 * LDSP + (sc1 << 2));
    async_cp16(g0 + sr2 * DD + (sc2 << 2), Bs + sr2 * LDSP + (sc2 << 2));
    async_cp16(g0 + sr3 * DD + (sc3 << 2), Bs + sr3 * LDSP + (sc3 << 2));
    __builtin_amdgcn_s_wait_asynccnt(0);
#else
    const float4* g = (const float4*)g0;
    float4 b0 = g[sr0 * (DD / 4) + sc0];
    float4 b1 = g[sr1 * (DD / 4) + sc1];
    float4 b2 = g[sr2 * (DD / 4) + sc2];
    float4 b3 = g[sr3 * (DD / 4) + sc3];
    *(float4*)(Bs + sr0 * LDSP + (sc0 << 2)) = b0;
    *(float4*)(Bs + sr1 * LDSP + (sc1 << 2)) = b1;
    *(float4*)(Bs + sr2 * LDSP + (sc2 << 2)) = b2;
    *(float4*)(Bs + sr3 * LDSP + (sc3 << 2)) = b3;
#endif
  }

  // This wave's row labels (rows rowBase + wave*16 + 8*lhalf + r).
  const int myRow0 = wave * 16 + 8 * lhalf;
  int labi[8];
#pragma unroll
  for (int r = 0; r < 8; ++r) labi[r] = labels[rowBase + myRow0 + r];

  float bpv[8], bnv[8];
  int bpi[8], bni[8];
#pragma unroll
  for (int r = 0; r < 8; ++r) { bpv[r] = -INF; bpi[r] = 0; bnv[r] = INF; bni[r] = 0; }

  const float* aptr = As + (wave * 16 + l16) * LDSP + 2 * lhalf;

  __syncthreads();

  for (int ct = 0; ct < TILES_PER_SPLIT; ++ct) {
    float* bufCur = Bs + (ct & 1) * (16 * LDSP);
    const int nct = ct + 1;
    float* bufNext = Bs + (nct & 1) * (16 * LDSP);

    // ---- Kick off the next tile's global->LDS transfer ----
#if HAS_ASYNC_LDS
    if (nct < TILES_PER_SPLIT) {
      const float* g = e + (size_t)(colBase0 + nct * 16) * DD;
      async_cp16(g + sr0 * DD + (sc0 << 2), bufNext + sr0 * LDSP + (sc0 << 2));
      async_cp16(g + sr1 * DD + (sc1 << 2), bufNext + sr1 * LDSP + (sc1 << 2));
      async_cp16(g + sr2 * DD + (sc2 << 2), bufNext + sr2 * LDSP + (sc2 << 2));
      async_cp16(g + sr3 * DD + (sc3 << 2), bufNext + sr3 * LDSP + (sc3 << 2));
    }
#else
    float4 n0, n1, n2, n3;
    if (nct < TILES_PER_SPLIT) {
      const float4* g = (const float4*)(e + (size_t)(colBase0 + nct * 16) * DD);
      n0 = g[sr0 * (DD / 4) + sc0];
      n1 = g[sr1 * (DD / 4) + sc1];
      n2 = g[sr2 * (DD / 4) + sc2];
      n3 = g[sr3 * (DD / 4) + sc3];
    }
#endif

    // ---- 64 x V_WMMA_F32_16X16X4_F32, 4 independent accumulator chains ----
    const float* bptr = bufCur + l16 * LDSP + 2 * lhalf;
    v8f c0 = {0.f,0.f,0.f,0.f,0.f,0.f,0.f,0.f};
    v8f c1 = c0, c2 = c0, c3 = c0;
#pragma unroll
    for (int ks = 0; ks < 16; ++ks) {
      const int k0 = ks * 16;  // 4 k-steps of 4
      v2f a0 = *(const v2f*)(aptr + k0 + 0),  b0 = *(const v2f*)(bptr + k0 + 0);
      v2f a1 = *(const v2f*)(aptr + k0 + 4),  b1 = *(const v2f*)(bptr + k0 + 4);
      v2f a2 = *(const v2f*)(aptr + k0 + 8),  b2 = *(const v2f*)(bptr + k0 + 8);
      v2f a3 = *(const v2f*)(aptr + k0 + 12), b3 = *(const v2f*)(bptr + k0 + 12);
      c0 = __builtin_amdgcn_wmma_f32_16x16x4_f32(false, a0, false, b0, (short)0, c0, false, false);
      c1 = __builtin_amdgcn_wmma_f32_16x16x4_f32(false, a1, false, b1, (short)0, c1, false, false);
      c2 = __builtin_amdgcn_wmma_f32_16x16x4_f32(false, a2, false, b2, (short)0, c2, false, false);
      c3 = __builtin_amdgcn_wmma_f32_16x16x4_f32(false, a3, false, b3, (short)0, c3, false, false);
    }

    // ---- Epilogue: lane covers column j = colBase + l16, rows myRow0 + r ----
    const int j = colBase0 + ct * 16 + l16;
    const float sqj = sq[j];
    const int labj = labels[j];
    const int iRow = rowBase + myRow0;
#pragma unroll
    for (int r = 0; r < 8; ++r) {
      const float d2 = sqj - 2.0f * (c0[r] + c1[r] + c2[r] + c3[r]);
      if (labj == labi[r]) {
        if (j != iRow + r && d2 > bpv[r]) { bpv[r] = d2; bpi[r] = j; }
      } else {
        if (d2 < bnv[r]) { bnv[r] = d2; bni[r] = j; }
      }
    }

    // ---- Complete the next tile's transfer, then one barrier per tile ----
#if HAS_ASYNC_LDS
    if (nct < TILES_PER_SPLIT) __builtin_amdgcn_s_wait_asynccnt(0);
#else
    if (nct < TILES_PER_SPLIT) {
      *(float4*)(bufNext + sr0 * LDSP + (sc0 << 2)) = n0;
      *(float4*)(bufNext + sr1 * LDSP + (sc1 << 2)) = n1;
      *(float4*)(bufNext + sr2 * LDSP + (sc2 << 2)) = n2;
      *(float4*)(bufNext + sr3 * LDSP + (sc3 << 2)) = n3;
    }
#endif
    __syncthreads();
  }

  // Butterfly reduce across the 16 lanes that share each row (first-index tiebreak).
#pragma unroll
  for (int m = 1; m < 16; m <<= 1) {
#pragma unroll
    for (int r = 0; r < 8; ++r) {
      float ov = __shfl_xor(bpv[r], m, 16); int oi = __shfl_xor(bpi[r], m, 16);
      if (ov > bpv[r] || (ov == bpv[r] && oi < bpi[r])) { bpv[r] = ov; bpi[r] = oi; }
      ov = __shfl_xor(bnv[r], m, 16); oi = __shfl_xor(bni[r], m, 16);
      if (ov < bnv[r] || (ov == bnv[r] && oi < bni[r])) { bnv[r] = ov; bni[r] = oi; }
    }
  }
  if (l16 == 0) {
#pragma unroll
    for (int r = 0; r < 8; ++r) {
      const int i = rowBase + myRow0 + r;
      posv[split * NN + i] = bpv[r]; posi[split * NN + i] = bpi[r];
      negv[split * NN + i] = bnv[r]; negi[split * NN + i] = bni[r];
    }
  }
}

// ---------------------------------------------------------------------------
// Kernel 3: merge the 4 column-split partials per row, then emit
// (embeds, embeds[pos], embeds[neg]) with b128 copies. grid=N, block=64.
// ---------------------------------------------------------------------------
__global__ __launch_bounds__(64) void tk_gather(
    const float* __restrict__ e,
    const float* __restrict__ posv, const int* __restrict__ posi,
    const float* __restrict__ negv, const int* __restrict__ negi,
    float* __restrict__ out) {
  const int i = blockIdx.x;
  const float INF = __builtin_inff();
  float bp = -INF, bn = INF;
  int bpi = 0, bni = 0;
#pragma unroll
  for (int s = 0; s < COL_SPLITS; ++s) {
    float v = posv[s * NN + i]; int id = posi[s * NN + i];
    if (v > bp || (v == bp && id < bpi)) { bp = v; bpi = id; }
    v = negv[s * NN + i]; id = negi[s * NN + i];
    if (v < bn || (v == bn && id < bni)) { bn = v; bni = id; }
  }
  const int t = threadIdx.x;
  const float4* e4 = (const float4*)e;
  float4* o4 = (float4*)out;
  const int P = DD / 4;          // 64
  const int nd4 = NN * P;
  o4[i * P + t]           = e4[(size_t)i * P + t];
  o4[nd4 + i * P + t]     = e4[(size_t)bpi * P + t];
  o4[2 * nd4 + i * P + t] = e4[(size_t)bni * P + t];
}

// ---------------------------------------------------------------------------
extern "C" void kernel_launch(void* const* d_in, const int* in_sizes, int n_in,
                              void* d_out, int out_size, void* d_ws, size_t ws_size,
                              hipStream_t stream) {
  (void)in_sizes; (void)n_in; (void)out_size; (void)ws_size;
  const float* embeds = (const float*)d_in[0];
  const int* labels = (const int*)d_in[1];
  float* out = (float*)d_out;

  char* ws = (char*)d_ws;
  float* sq   = (float*)ws;                                   // N floats
  float* posv = (float*)(ws + (size_t)NN * 4);                // 4*N floats
  int*   posi = (int*)  (ws + (size_t)NN * 4 * 5);            // 4*N ints
  float* negv = (float*)(ws + (size_t)NN * 4 * 9);            // 4*N floats
  int*   negi = (int*)  (ws + (size_t)NN * 4 * 13);           // 4*N ints

  tk_sqnorm<<<NN / 8, 256, 0, stream>>>(embeds, sq);

  const size_t shmem = (size_t)(ROWS_PER_BLOCK + 32) * LDSP * sizeof(float); // ~163 KB of 320 KB WGP LDS
  (void)hipFuncSetAttribute((const void*)tk_hardmine,
                            hipFuncAttributeMaxDynamicSharedMemorySize, (int)shmem);
  dim3 grid(NN / ROWS_PER_BLOCK, COL_SPLITS);
  tk_hardmine<<<grid, 256, shmem, stream>>>(embeds, labels, sq,
                                            posv, posi, negv, negi);

  tk_gather<<<NN, 64, 0, stream>>>(embeds, posv, posi, negv, negi, out);
}